// PointNetPlus_15788299780188
// MI455X (gfx1250) — compile-verified
//
#include <hip/hip_runtime.h>
#include <hip/hip_bf16.h>
#include <math.h>

typedef __attribute__((ext_vector_type(2))) float v2f;
typedef __attribute__((ext_vector_type(4))) float v4f;
typedef __attribute__((ext_vector_type(8))) float v8f;

#define EPS_BN 1e-5f
constexpr int B_   = 8;
constexpr int N_   = 2048;
constexpr int KNN_ = 20;
constexpr int NT_  = N_ / 16;   // 128 point-tiles
constexpr int CT_  = 512;       // channels in concatenated feature buffer

// ---------------------------------------------------------------------------
// 1) KNN: per-point top-20 nearest neighbors (neg sq-euclid top-k, self incl.)
// ---------------------------------------------------------------------------
__global__ void knn_kernel(const float* __restrict__ x, int* __restrict__ idx) {
  __shared__ float px[N_], py[N_], pz[N_];
  int b     = blockIdx.x / (N_ / 256);
  int chunk = blockIdx.x % (N_ / 256);
  const float* xb = x + (long)b * 3 * N_;
  for (int t = threadIdx.x; t < N_; t += 256) {
    px[t] = xb[t];
    py[t] = xb[N_ + t];
    pz[t] = xb[2 * N_ + t];
  }
  __syncthreads();
  int i = chunk * 256 + threadIdx.x;
  float qx = px[i], qy = py[i], qz = pz[i];
  float db[KNN_];
  int   ib[KNN_];
#pragma unroll
  for (int k = 0; k < KNN_; ++k) { db[k] = 3.4e38f; ib[k] = 0; }
  for (int j = 0; j < N_; ++j) {
    float dx = qx - px[j], dy = qy - py[j], dz = qz - pz[j];
    float d = dx * dx + dy * dy + dz * dz;
    if (d < db[KNN_ - 1]) {              // strict <: stable ties like lax.top_k
      int p = KNN_ - 1;
      while (p > 0 && d < db[p - 1]) { db[p] = db[p - 1]; ib[p] = ib[p - 1]; --p; }
      db[p] = d; ib[p] = j;
    }
  }
  int* ob = idx + ((long)b * N_ + i) * KNN_;
#pragma unroll
  for (int k = 0; k < KNN_; ++k) ob[k] = ib[k];
}

// ---------------------------------------------------------------------------
// 2) EdgeConv: out[b,n,o] = max_k lrelu(bn(sum_c W[o,c]*in[b, idx[b,n,k], c]))
//    One wave per 16(out-ch) x 16(points) tile, f32 WMMA 16x16x4 over C_in.
//    NMAJOR: input stored point-major (row stride in_rs); else channel-major.
//    Output always point-major into the 512-ch cat buffer (contiguous v4f).
// ---------------------------------------------------------------------------
template <int CIN, bool NMAJOR>
__global__ void edgeconv_wmma(const float* __restrict__ in, long in_bs, int in_rs,
                              const float* __restrict__ W,
                              const float* __restrict__ g, const float* __restrict__ bb,
                              const int* __restrict__ idx,
                              float* __restrict__ out, long out_bs,
                              int o_tiles) {
  constexpr int  NCH = (CIN + 3) / 4;
  constexpr bool PAD = (CIN % 4) != 0;

  int wave = blockIdx.x * (blockDim.x >> 5) + (threadIdx.x >> 5);
  int lane = threadIdx.x & 31;
  int half = lane >> 4;       // 0: K rows 0,1 / M rows 0-7 ; 1: K rows 2,3 / M rows 8-15
  int lrow = lane & 15;

  int ntile = wave % NT_;
  int otile = (wave / NT_) % o_tiles;
  int b     = wave / (NT_ * o_tiles);
  int n0 = ntile * 16, o0 = otile * 16;

  // Preload the A-matrix (weights) in WMMA layout
  v2f wr[NCH];
#pragma unroll
  for (int cc = 0; cc < NCH; ++cc) {
    int c = 4 * cc + 2 * half;
    if (PAD) {
      wr[cc].x = (c     < CIN) ? W[(long)(o0 + lrow) * CIN + c]     : 0.f;
      wr[cc].y = (c + 1 < CIN) ? W[(long)(o0 + lrow) * CIN + c + 1] : 0.f;
    } else {
      wr[cc] = *(const v2f*)(W + (long)(o0 + lrow) * CIN + c);   // 8B aligned
    }
  }

  // BN scale/bias per accumulator row
  const float inv = rsqrtf(1.f + EPS_BN);
  float sc[8], bi[8];
#pragma unroll
  for (int r = 0; r < 8; ++r) {
    int o = o0 + r + 8 * half;
    sc[r] = g[o] * inv;
    bi[r] = bb[o];
  }

  const float* inb  = in + (long)b * in_bs;
  const int*   idxb = idx + ((long)b * N_ + n0 + lrow) * KNN_;

  v8f vmax;
#pragma unroll
  for (int r = 0; r < 8; ++r) vmax[r] = -3.4e38f;

  for (int k = 0; k < KNN_; ++k) {
    int j = idxb[k];                     // lanes l and l+16 read the same index
    const float* pj = NMAJOR ? (inb + (long)j * in_rs) : nullptr;
    v8f acc = {};
#pragma unroll
    for (int cc = 0; cc < NCH; ++cc) {
      int c = 4 * cc + 2 * half;
      v2f bv;
      if (NMAJOR) {
        bv = *(const v2f*)(pj + c);      // gathered point's channels contiguous
      } else {
        bv.x = (!PAD || c     < CIN) ? inb[(long)c * N_ + j]       : 0.f;
        bv.y = (!PAD || c + 1 < CIN) ? inb[(long)(c + 1) * N_ + j] : 0.f;
      }
      acc = __builtin_amdgcn_wmma_f32_16x16x4_f32(false, wr[cc], false, bv,
                                                  (short)0, acc, false, false);
    }
#pragma unroll
    for (int r = 0; r < 8; ++r) {
      float v = acc[r] * sc[r] + bi[r];
      v = (v >= 0.f) ? v : 0.2f * v;
      vmax[r] = fmaxf(vmax[r], v);
    }
  }

  // point-major store: 8 contiguous channels per lane -> two b128 stores
  float* ob = out + (long)b * out_bs + (long)(n0 + lrow) * CT_ + o0 + 8 * half;
  v4f lo, hi;
  lo.x = vmax[0]; lo.y = vmax[1]; lo.z = vmax[2]; lo.w = vmax[3];
  hi.x = vmax[4]; hi.y = vmax[5]; hi.z = vmax[6]; hi.w = vmax[7];
  *(v4f*)(ob)     = lo;
  *(v4f*)(ob + 4) = hi;
}

// ---------------------------------------------------------------------------
// 3) conv5 (512 -> 1024) + BN + lrelu + in-wave tile pooling (deterministic)
// ---------------------------------------------------------------------------
__global__ void conv5_pool_wmma(const float* __restrict__ cat, const float* __restrict__ W5,
                                const float* __restrict__ g, const float* __restrict__ bb,
                                float* __restrict__ pmax, float* __restrict__ psum) {
  int wave = blockIdx.x * (blockDim.x >> 5) + (threadIdx.x >> 5);
  int lane = threadIdx.x & 31;
  int half = lane >> 4;
  int lrow = lane & 15;

  int ntile = wave % NT_;
  int otile = (wave / NT_) % 64;       // 1024/16
  int b     = wave / (NT_ * 64);
  int n0 = ntile * 16, o0 = otile * 16;
  int n = n0 + lrow;

  const float* inb  = cat + ((long)b * N_ + n) * CT_;           // point-major row
  const float* wrow = W5 + (long)(o0 + lrow) * CT_ + 2 * half;

  v8f acc = {};
#pragma unroll 8
  for (int cc = 0; cc < 128; ++cc) {
    int c = 4 * cc + 2 * half;
    v2f av = *(const v2f*)(wrow + 4 * cc);
    v2f bv = *(const v2f*)(inb + c);
    acc = __builtin_amdgcn_wmma_f32_16x16x4_f32(false, av, false, bv,
                                                (short)0, acc, false, false);
  }

  const float inv = rsqrtf(1.f + EPS_BN);
  float m[8], s[8];
#pragma unroll
  for (int r = 0; r < 8; ++r) {
    int o = o0 + r + 8 * half;
    float v = acc[r] * g[o] * inv + bb[o];
    v = (v >= 0.f) ? v : 0.2f * v;
    m[r] = v;
    s[r] = v;
  }
  // butterfly reduction over the 16 points of this tile (within each half-wave)
#pragma unroll
  for (int off = 1; off < 16; off <<= 1) {
#pragma unroll
    for (int r = 0; r < 8; ++r) {
      m[r] = fmaxf(m[r], __shfl_xor(m[r], off, 32));
      s[r] += __shfl_xor(s[r], off, 32);
    }
  }
  if (lrow == 0) {
#pragma unroll
    for (int r = 0; r < 8; ++r) {
      int o = o0 + r + 8 * half;
      pmax[((long)b * 1024 + o) * NT_ + ntile] = m[r];
      psum[((long)b * 1024 + o) * NT_ + ntile] = s[r];
    }
  }
}

__global__ void pool_reduce_kernel(const float* __restrict__ pmax,
                                   const float* __restrict__ psum,
                                   float* __restrict__ fvec) {
  int id = blockIdx.x * blockDim.x + threadIdx.x;   // B_*1024
  if (id >= B_ * 1024) return;
  int b = id / 1024, o = id % 1024;
  const float* pm = pmax + (long)id * NT_;
  const float* ps = psum + (long)id * NT_;
  float m = -3.4e38f, s = 0.f;
  for (int t = 0; t < NT_; ++t) { m = fmaxf(m, pm[t]); s += ps[t]; }
  fvec[(long)b * 2048 + o]        = m;
  fvec[(long)b * 2048 + 1024 + o] = s * (1.f / (float)N_);
}

// ---------------------------------------------------------------------------
// 4) tiny FC chain (thread-per-output dot products)
// ---------------------------------------------------------------------------
__global__ void fc_kernel(const float* __restrict__ in, const float* __restrict__ W,
                          const float* __restrict__ lb, const float* __restrict__ g,
                          const float* __restrict__ bb, float* __restrict__ out,
                          int IN, int OUT, int act) {
  int id = blockIdx.x * blockDim.x + threadIdx.x;
  if (id >= B_ * OUT) return;
  int b = id / OUT, o = id % OUT;
  const float* iv = in + (long)b * IN;
  const float* wv = W + (long)o * IN;
  float s = 0.f;
  for (int i = 0; i < IN; ++i) s = fmaf(iv[i], wv[i], s);
  if (lb) s += lb[o];
  if (g)  s = s * g[o] * rsqrtf(1.f + EPS_BN) + bb[o];
  if (act) s = (s >= 0.f) ? s : 0.2f * s;
  out[(long)b * OUT + o] = s;
}

// ---------------------------------------------------------------------------
// Host launch
// ---------------------------------------------------------------------------
extern "C" void kernel_launch(void* const* d_in, const int* in_sizes, int n_in,
                              void* d_out, int out_size, void* d_ws, size_t ws_size,
                              hipStream_t stream) {
  (void)in_sizes; (void)n_in; (void)out_size; (void)ws_size;

  const float* x   = (const float*)d_in[0];
  const float* W1  = (const float*)d_in[1];
  const float* W2  = (const float*)d_in[2];
  const float* W3  = (const float*)d_in[3];
  const float* W4  = (const float*)d_in[4];
  const float* W5  = (const float*)d_in[5];
  const float* g1  = (const float*)d_in[6],  *b1 = (const float*)d_in[7];
  const float* g2  = (const float*)d_in[8],  *b2 = (const float*)d_in[9];
  const float* g3  = (const float*)d_in[10], *b3 = (const float*)d_in[11];
  const float* g4  = (const float*)d_in[12], *b4 = (const float*)d_in[13];
  const float* g5  = (const float*)d_in[14], *b5 = (const float*)d_in[15];
  const float* g6  = (const float*)d_in[16], *b6 = (const float*)d_in[17];
  const float* g7  = (const float*)d_in[18], *b7 = (const float*)d_in[19];
  const float* L1W = (const float*)d_in[20];
  const float* L2W = (const float*)d_in[21];
  const float* L2b = (const float*)d_in[22];
  const float* L3W = (const float*)d_in[23];
  const float* L3b = (const float*)d_in[24];

  // workspace layout
  char*  ws     = (char*)d_ws;
  int*   idx_ws = (int*)ws;                                   // 8*2048*20 ints
  float* cat    = (float*)(ws + (size_t)B_ * N_ * KNN_ * 4);  // (B, N, 512) f32
  float* pmax   = cat + (size_t)B_ * N_ * CT_;                // (B, 1024, NT_)
  float* psum   = pmax + (size_t)B_ * 1024 * NT_;             // (B, 1024, NT_)
  float* fvec   = psum + (size_t)B_ * 1024 * NT_;             // (B, 2048)
  float* f1     = fvec + (size_t)B_ * 2048;                   // (B, 512)
  float* f2     = f1 + (size_t)B_ * 512;                      // (B, 256)

  const long catBS = (long)N_ * CT_;

  // 1) KNN
  knn_kernel<<<B_ * (N_ / 256), 256, 0, stream>>>(x, idx_ws);

  // 2) EdgeConv chain; outputs interleaved point-major in `cat`:
  //    x1 -> cat[:, :, 0:64), x2 -> [64:128), x3 -> [128:256), x4 -> [256:512)
  edgeconv_wmma<3, false><<<(B_ * 4 * NT_) / 4, 128, 0, stream>>>(
      x, (long)3 * N_, 0, W1, g1, b1, idx_ws, cat, catBS, 4);
  edgeconv_wmma<64, true><<<(B_ * 4 * NT_) / 4, 128, 0, stream>>>(
      cat, catBS, CT_, W2, g2, b2, idx_ws, cat + 64, catBS, 4);
  edgeconv_wmma<64, true><<<(B_ * 8 * NT_) / 4, 128, 0, stream>>>(
      cat + 64, catBS, CT_, W3, g3, b3, idx_ws, cat + 128, catBS, 8);
  edgeconv_wmma<128, true><<<(B_ * 16 * NT_) / 4, 128, 0, stream>>>(
      cat + 128, catBS, CT_, W4, g4, b4, idx_ws, cat + 256, catBS, 16);

  // 3) conv5 + deterministic fused pooling
  conv5_pool_wmma<<<(B_ * 64 * NT_) / 4, 128, 0, stream>>>(cat, W5, g5, b5, pmax, psum);
  pool_reduce_kernel<<<(B_ * 1024 + 255) / 256, 256, 0, stream>>>(pmax, psum, fvec);

  // 4) FC head
  fc_kernel<<<(B_ * 512 + 255) / 256, 256, 0, stream>>>(
      fvec, L1W, nullptr, g6, b6, f1, 2048, 512, 1);
  fc_kernel<<<(B_ * 256 + 255) / 256, 256, 0, stream>>>(
      f1, L2W, L2b, g7, b7, f2, 512, 256, 1);
  fc_kernel<<<(B_ * 40 + 255) / 256, 256, 0, stream>>>(
      f2, L3W, L3b, nullptr, nullptr, (float*)d_out, 256, 40, 0);
}